// AnaphoricityScorer_4294967296451
// MI455X (gfx1250) — compile-verified
//
#include <hip/hip_runtime.h>
#include <hip/hip_bf16.h>

typedef __attribute__((ext_vector_type(2))) float v2f;
typedef __attribute__((ext_vector_type(8))) float v8f;

#define EMB     1024
#define NANT    50
#define BATCH   1024
#define PWE     64
#define KTOT    3136      // 3*EMB + PWE
#define HID     1024
#define TM      64        // rows of pair-matrix per block
#define TN      128       // hidden cols per pass
#define KC      64        // K chunk
#define NCHUNKS (HID / TN)   // 8
#define KCHUNKS (KTOT / KC)  // 49
#define BSTRIDE (2 * TN + 4) // 260 floats: 1040B row, 16B-aligned, padded
#define EPS_VAL 1e-7f

// ---------------------------------------------------------------------------
// Fused kernel: on-the-fly pair construction -> f32 WMMA GEMM (K=3136) ->
// bias + leaky_relu -> dot with W_out -> + rough + b_out -> scatter to out.
// B tile stored K-pair-interleaved in LDS so each B fragment is one b64 load.
// ---------------------------------------------------------------------------
__global__ __launch_bounds__(256) void anaph_scorer_kernel(
    const float* __restrict__ am,      // all_mentions      [8192,1024]
    const float* __restrict__ ment,    // mentions_batch    [1024,1024]
    const float* __restrict__ pw,      // pw_batch          [1024,50,64]
    const float* __restrict__ rough,   // rough scores      [1024,50]
    const float* __restrict__ W1,      // [3136,1024]
    const float* __restrict__ b1,      // [1024]
    const float* __restrict__ Wout,    // [1024]
    const float* __restrict__ bout,    // [1]
    const int*   __restrict__ topIdx,  // [1024,50]
    float* __restrict__ out)           // [1024,51]
{
    __shared__ float As[TM][68];            // 64x64 pair tile (17,408 B)
    __shared__ float Bs[KC / 2][BSTRIDE];   // interleaved W1 tile (33,280 B)
    __shared__ int   rowBs[TM];             // batch index per row
    __shared__ int   rowGs[TM];             // gathered mention index per row
    __shared__ float partial[TM];           // second-layer row sums

    const int tid  = threadIdx.x;
    const int lane = tid & 31;
    const int w    = tid >> 5;     // wave id 0..7
    const int rt   = w & 3;        // row tile 0..3 (16 rows each)
    const int cg   = w >> 2;       // col group 0/1 -> col tiles cg*4 .. cg*4+3
    const int half = lane >> 4;    // lane half (K split per ISA frag layout)
    const int lr   = lane & 15;
    const int m0   = blockIdx.x * TM;

    if (tid < TM) {
        int m = m0 + tid;
        int b = m / NANT;
        int n = m - b * NANT;
        rowBs[tid]   = b;
        rowGs[tid]   = topIdx[b * NANT + n];
        partial[tid] = 0.0f;
    }
    __syncthreads();

    float rowsum[8];
#pragma unroll
    for (int e = 0; e < 8; ++e) rowsum[e] = 0.0f;

    for (int nc = 0; nc < NCHUNKS; ++nc) {
        const int n0 = nc * TN;
        v8f acc[4];
        const v8f zero = {};
#pragma unroll
        for (int t = 0; t < 4; ++t) acc[t] = zero;

        for (int kc = 0; kc < KCHUNKS; ++kc) {
            const int k0     = kc * KC;
            const int region = kc >> 4;       // 0,1,2 emb blocks; 3 = pw
            __syncthreads();                   // LDS reuse guard

            // ---- build A tile (pair matrix rows): 4 float4 per thread ----
            {
                const int k = (tid & 15) * 4;  // 0..60
#pragma unroll
                for (int p = 0; p < 4; ++p) {
                    const int row = p * 16 + (tid >> 4);
                    const int b = rowBs[row];
                    const int g = rowGs[row];
                    float4 v;
                    if (region == 0) {
                        v = *(const float4*)(ment + (size_t)b * EMB + k0 + k);
                    } else if (region == 1) {
                        v = *(const float4*)(am + (size_t)g * EMB + (k0 - EMB) + k);
                    } else if (region == 2) {
                        float4 x = *(const float4*)(ment + (size_t)b * EMB + (k0 - 2 * EMB) + k);
                        float4 y = *(const float4*)(am + (size_t)g * EMB + (k0 - 2 * EMB) + k);
                        v = make_float4(x.x * y.x, x.y * y.y, x.z * y.z, x.w * y.w);
                    } else {
                        v = *(const float4*)(pw + (size_t)(m0 + row) * PWE + k);
                    }
                    *(float4*)&As[row][k] = v;
                }
            }
            // ---- load B tile (W1 chunk), K-pair interleaved ----
            // Bs[kk][c*2 + j] = W1[k0 + 2*kk + j][n0 + c]
            {
                const int c4 = (tid & 31) * 4;  // 0..124
#pragma unroll
                for (int p = 0; p < 4; ++p) {
                    const int kk = p * 8 + (tid >> 5);         // 0..31 (row pair)
                    const float* src = W1 + (size_t)(k0 + 2 * kk) * HID + n0 + c4;
                    float4 x = *(const float4*)(src);
                    float4 y = *(const float4*)(src + HID);
                    *(float4*)&Bs[kk][c4 * 2]     = make_float4(x.x, y.x, x.y, y.y);
                    *(float4*)&Bs[kk][c4 * 2 + 4] = make_float4(x.z, y.z, x.w, y.w);
                }
            }
            __syncthreads();

            // ---- 16 k-steps of v_wmma_f32_16x16x4_f32 per col tile ----
#pragma unroll
            for (int s = 0; s < 16; ++s) {
                const int kb = s * 4;
                // A frag: lane l -> row l%16, K = kb + 2*(l/16) .. +1
                v2f a = *(const v2f*)&As[rt * 16 + lr][kb + 2 * half];
                const int kk = 2 * s + half;    // interleaved row-pair index
#pragma unroll
                for (int t = 0; t < 4; ++t) {
                    const int ct = (cg * 4 + t) * 16 + lr;  // local col in B tile
                    v2f bf = *(const v2f*)&Bs[kk][ct * 2];  // one ds_load_b64
                    acc[t] = __builtin_amdgcn_wmma_f32_16x16x4_f32(
                        false, a, false, bf, (short)0, acc[t], false, false);
                }
            }
        }

        // ---- fused layer 2: bias + leaky_relu + dot with W_out ----
#pragma unroll
        for (int t = 0; t < 4; ++t) {
            const int col = n0 + (cg * 4 + t) * 16 + lr;
            const float bb = b1[col];
            const float wo = Wout[col];
#pragma unroll
            for (int e = 0; e < 8; ++e) {
                float v = acc[t][e] + bb;
                v = (v > 0.0f) ? v : 0.01f * v;
                rowsum[e] += v * wo;
            }
        }
    }

    // reduce each row partial over the 16 lanes of its half (wave32 butterflies)
#pragma unroll
    for (int e = 0; e < 8; ++e) {
        float v = rowsum[e];
        v += __shfl_xor(v, 1, 32);
        v += __shfl_xor(v, 2, 32);
        v += __shfl_xor(v, 4, 32);
        v += __shfl_xor(v, 8, 32);
        rowsum[e] = v;
    }
    if (lr == 0) {
#pragma unroll
        for (int e = 0; e < 8; ++e)
            atomicAdd(&partial[rt * 16 + 8 * half + e], rowsum[e]);  // ds_add_f32
    }
    __syncthreads();

    if (tid < TM) {
        int m = m0 + tid;
        int b = m / NANT;
        int n = m - b * NANT;
        out[b * (NANT + 1) + 1 + n] = rough[m] + partial[tid] + bout[0];
    }
}

// Dummy EPS column: out[b, 0] = 1e-7
__global__ void dummy_col_kernel(float* __restrict__ out) {
    int b = blockIdx.x * blockDim.x + threadIdx.x;
    if (b < BATCH) out[b * (NANT + 1)] = EPS_VAL;
}

extern "C" void kernel_launch(void* const* d_in, const int* in_sizes, int n_in,
                              void* d_out, int out_size, void* d_ws, size_t ws_size,
                              hipStream_t stream) {
    const float* am    = (const float*)d_in[0];
    const float* ment  = (const float*)d_in[1];
    const float* pw    = (const float*)d_in[2];
    const float* rough = (const float*)d_in[3];
    const float* W1    = (const float*)d_in[4];
    const float* b1    = (const float*)d_in[5];
    const float* Wout  = (const float*)d_in[6];
    const float* bout  = (const float*)d_in[7];
    const int*   idx   = (const int*)d_in[8];
    float* out = (float*)d_out;

    dummy_col_kernel<<<(BATCH + 255) / 256, 256, 0, stream>>>(out);

    const int M = BATCH * NANT;  // 51200 rows, TM=64 -> 800 blocks
    anaph_scorer_kernel<<<M / TM, 256, 0, stream>>>(
        am, ment, pw, rough, W1, b1, Wout, bout, idx, out);
}